// RegionOfInterest_3255585210658
// MI455X (gfx1250) — compile-verified
//
#include <hip/hip_runtime.h>

// ---------------------------------------------------------------------------
// RegionOfInterest (crop_and_resize, bilinear) for MI455X / gfx1250.
//
// Bandwidth-bound op (~114 MB effective HBM traffic; 64MB image stays resident
// in the 192MB L2 across ~196MB of gathered reads). The 4-tap bilinear blend
// over 16-channel tiles is expressed as V_WMMA_F32_16X16X4_F32:
//     D(16x16) = A(16x4 weights) x B(4x16 corner pixels)
// One wave32 per (box, gy, gx) position; 16 WMMAs cover all 256 channels.
// Output stores are non-temporal (TH=NT): written once, never re-read, so
// they must not evict the L2-resident image.
// ---------------------------------------------------------------------------

typedef __attribute__((ext_vector_type(2))) float v2f;
typedef __attribute__((ext_vector_type(8))) float v8f;

#define H_IMG 256
#define W_IMG 256
#define C_IMG 256
#define N_BOXES 1000
#define GH 7
#define GW 7
#define NPOS (N_BOXES * GH * GW)   // 49000 positions; 8 waves/block -> 6125 blocks, no tail

__global__ __launch_bounds__(256) void roi_wmma_kernel(
    const float* __restrict__ meta,
    const float* __restrict__ img,
    const float* __restrict__ boxes,
    float* __restrict__ out)
{
    const int lane = threadIdx.x & 31;
    const int waveInBlock = threadIdx.x >> 5;
    const int pos = blockIdx.x * 8 + waveInBlock;     // wave-uniform
    if (pos >= NPOS) return;                          // uniform per wave: EXEC stays all-1s

    const int box = pos / (GH * GW);
    const int rem = pos - box * (GH * GW);
    const int gy = rem / GW;
    const int gx = rem - gy * GW;

    // ---- box -> normalized -> image-space coordinates (mirrors reference) ----
    const float h = meta[0];                // 256.0
    const float w = meta[1];                // 256.0
    const float bx1 = boxes[box * 4 + 0];
    const float by1 = boxes[box * 4 + 1];
    const float bx2 = boxes[box * 4 + 2];
    const float by2 = boxes[box * 4 + 3];

    const float y1n = by1 / (h - 1.0f);
    const float x1n = bx1 / (w - 1.0f);
    const float y2n = by2 / (h - 1.0f);
    const float x2n = bx2 / (w - 1.0f);

    const float ty = (float)gy / (float)(GH - 1);
    const float tx = (float)gx / (float)(GW - 1);
    const float in_y = (y1n + ty * (y2n - y1n)) * (float)(H_IMG - 1);
    const float in_x = (x1n + tx * (x2n - x1n)) * (float)(W_IMG - 1);

    const bool valid = (in_y >= 0.0f) && (in_y <= (float)(H_IMG - 1)) &&
                       (in_x >= 0.0f) && (in_x <= (float)(W_IMG - 1));

    const float y0f = floorf(in_y);
    const float x0f = floorf(in_x);
    const float ly = in_y - y0f;
    const float lx = in_x - x0f;

    const int y0  = (int)fminf(fmaxf(y0f,        0.0f), (float)(H_IMG - 1));
    const int y1i = (int)fminf(fmaxf(y0f + 1.0f, 0.0f), (float)(H_IMG - 1));
    const int x0  = (int)fminf(fmaxf(x0f,        0.0f), (float)(W_IMG - 1));
    const int x1i = (int)fminf(fmaxf(x0f + 1.0f, 0.0f), (float)(W_IMG - 1));

    // Bilinear weights, K-order matches B rows: K0=g00, K1=g01, K2=g10, K3=g11
    float w00 = (1.0f - ly) * (1.0f - lx);
    float w01 = (1.0f - ly) * lx;
    float w10 = ly * (1.0f - lx);
    float w11 = ly * lx;
    if (!valid) { w00 = 0.0f; w01 = 0.0f; w10 = 0.0f; w11 = 0.0f; }

    // Corner pixel channel vectors (256 fp32 each)
    const float* p00 = img + (size_t)(y0  * W_IMG + x0 ) * C_IMG;
    const float* p01 = img + (size_t)(y0  * W_IMG + x1i) * C_IMG;
    const float* p10 = img + (size_t)(y1i * W_IMG + x0 ) * C_IMG;
    const float* p11 = img + (size_t)(y1i * W_IMG + x1i) * C_IMG;

    const int half = lane >> 4;      // 0: lanes 0-15 (K0/K1), 1: lanes 16-31 (K2/K3)
    const int cn   = lane & 15;      // channel-within-tile / N index

    // ---- A matrix (16x4 f32), rows replicated with this position's weights ----
    // ISA layout: lanes 0-15 -> VGPR0=K0, VGPR1=K1 ; lanes 16-31 -> VGPR0=K2, VGPR1=K3
    v2f a;
    a.x = half ? w10 : w00;
    a.y = half ? w11 : w01;

    // ---- B matrix lane base pointers ----
    // B VGPR0: K0 row (g00) on lanes 0-15, K2 row (g10) on lanes 16-31
    // B VGPR1: K1 row (g01) on lanes 0-15, K3 row (g11) on lanes 16-31
    const float* pb0 = (half ? p10 : p00) + cn;
    const float* pb1 = (half ? p11 : p01) + cn;

    float* op = out + (size_t)pos * C_IMG;   // out[box][gy][gx][:]

    const v8f cz = {0.f, 0.f, 0.f, 0.f, 0.f, 0.f, 0.f, 0.f};

    // 16 channel tiles of 16; process in pairs so every store is a full-wave
    // coalesced 128B non-temporal global_store_b32.
    #pragma unroll
    for (int t = 0; t < 16; t += 2) {
        v2f bE, bO;                         // B for even / odd tile
        bE.x = pb0[(t    ) * 16];
        bE.y = pb1[(t    ) * 16];
        bO.x = pb0[(t + 1) * 16];
        bO.y = pb1[(t + 1) * 16];

        // D = A x B + 0   (8 args: neg_a, A, neg_b, B, c_mod, C, reuse_a, reuse_b)
        v8f dE = __builtin_amdgcn_wmma_f32_16x16x4_f32(
            false, a, false, bE, (short)0, cz, false, false);
        v8f dO = __builtin_amdgcn_wmma_f32_16x16x4_f32(
            false, a, false, bO, (short)0, cz, false, false);

        // D VGPR0: lanes 0-15 = row M=0 (N=lane), lanes 16-31 = row M=8 (N=lane-16).
        // All A rows are identical, so both halves carry the channel result.
        float v = half ? dO[0] : dE[0];
        // Non-temporal: output is streamed once; don't evict L2-resident image.
        __builtin_nontemporal_store(v, &op[t * 16 + lane]);
    }
}

extern "C" void kernel_launch(void* const* d_in, const int* in_sizes, int n_in,
                              void* d_out, int out_size, void* d_ws, size_t ws_size,
                              hipStream_t stream) {
    (void)in_sizes; (void)n_in; (void)out_size; (void)d_ws; (void)ws_size;
    const float* meta  = (const float*)d_in[0];   // (1,3)  [H, W, 1]
    const float* image = (const float*)d_in[1];   // (1,256,256,256) f32, NHWC
    const float* boxes = (const float*)d_in[2];   // (1,1000,4) f32 [x1,y1,x2,y2]
    float* out = (float*)d_out;                   // (1,1000,7,7,256) f32

    const int blocks = NPOS / 8;                  // 6125, exact (8 waves/block)
    roi_wmma_kernel<<<blocks, 256, 0, stream>>>(meta, image, boxes, out);
}